// ElementwiseTensorProd_o1_9062380994710
// MI455X (gfx1250) — compile-verified
//
#include <hip/hip_runtime.h>

// Elementwise irrep tensor product, l=1 outputs, paths [(0,1,1),(1,0,1),(1,1,1)].
// Shapes (f32): z0l (N,256), z1l (N,3,256), z0r (N,256), z1r (N,3,256)
// Output: (N, 3, 3*256) = concat[p01, p10, cross/sqrt2] along channels.
//
// Memory-bound streaming kernel: 1 thread per (n, 16B rank-chunk).
// RANK=256 -> 64 float4 chunks per row; a wave32 covers 512 contiguous bytes
// per vector memory instruction (fully coalesced B128).

typedef float v4f __attribute__((ext_vector_type(4)));

#define RANK      256
#define R4        (RANK / 4)       // 64 float4 chunks per rank-row
#define NPATH     3
#define INV_SQRT2 0.70710678118654752440f

__global__ __launch_bounds__(256) void etp_o1_stream_kernel(
    const v4f* __restrict__ z0l,
    const v4f* __restrict__ z1l,
    const v4f* __restrict__ z0r,
    const v4f* __restrict__ z1r,
    v4f* __restrict__ out,
    int total)                      // N * R4
{
    const int t = blockIdx.x * blockDim.x + threadIdx.x;
    if (t >= total) return;

    const int n  = t >> 6;          // t / R4
    const int r4 = t & (R4 - 1);    // chunk within the rank row

    // ---- loads: 8 x global_load_b128, non-temporal (stream-once data) ----
    const v4f sl = __builtin_nontemporal_load(&z0l[n * R4 + r4]);
    const v4f sr = __builtin_nontemporal_load(&z0r[n * R4 + r4]);

    const int b1 = n * (3 * R4) + r4;            // base into (N,3,R4) arrays
    const v4f vl0 = __builtin_nontemporal_load(&z1l[b1 + 0 * R4]);
    const v4f vl1 = __builtin_nontemporal_load(&z1l[b1 + 1 * R4]);
    const v4f vl2 = __builtin_nontemporal_load(&z1l[b1 + 2 * R4]);
    const v4f vr0 = __builtin_nontemporal_load(&z1r[b1 + 0 * R4]);
    const v4f vr1 = __builtin_nontemporal_load(&z1r[b1 + 1 * R4]);
    const v4f vr2 = __builtin_nontemporal_load(&z1r[b1 + 2 * R4]);

    // ---- math (pure VALU, packed f32 x4) ----
    // path (0,1,1): scalar_l * vector_r
    const v4f p01_0 = sl * vr0;
    const v4f p01_1 = sl * vr1;
    const v4f p01_2 = sl * vr2;
    // path (1,0,1): vector_l * scalar_r
    const v4f p10_0 = vl0 * sr;
    const v4f p10_1 = vl1 * sr;
    const v4f p10_2 = vl2 * sr;
    // path (1,1,1): cross(vector_l, vector_r) / sqrt(2)
    const v4f cx = (vl1 * vr2 - vl2 * vr1) * INV_SQRT2;
    const v4f cy = (vl2 * vr0 - vl0 * vr2) * INV_SQRT2;
    const v4f cz = (vl0 * vr1 - vl1 * vr0) * INV_SQRT2;

    // ---- stores: 9 x global_store_b128, non-temporal ----
    // out float4 index: n*(3*NPATH*R4) + c*(NPATH*R4) + path*R4 + r4
    const int ob = n * (3 * NPATH * R4) + r4;
    __builtin_nontemporal_store(p01_0, &out[ob + 0 * (NPATH * R4) + 0 * R4]);
    __builtin_nontemporal_store(p10_0, &out[ob + 0 * (NPATH * R4) + 1 * R4]);
    __builtin_nontemporal_store(cx,    &out[ob + 0 * (NPATH * R4) + 2 * R4]);

    __builtin_nontemporal_store(p01_1, &out[ob + 1 * (NPATH * R4) + 0 * R4]);
    __builtin_nontemporal_store(p10_1, &out[ob + 1 * (NPATH * R4) + 1 * R4]);
    __builtin_nontemporal_store(cy,    &out[ob + 1 * (NPATH * R4) + 2 * R4]);

    __builtin_nontemporal_store(p01_2, &out[ob + 2 * (NPATH * R4) + 0 * R4]);
    __builtin_nontemporal_store(p10_2, &out[ob + 2 * (NPATH * R4) + 1 * R4]);
    __builtin_nontemporal_store(cz,    &out[ob + 2 * (NPATH * R4) + 2 * R4]);
}

extern "C" void kernel_launch(void* const* d_in, const int* in_sizes, int n_in,
                              void* d_out, int out_size, void* d_ws, size_t ws_size,
                              hipStream_t stream) {
    (void)n_in; (void)out_size; (void)d_ws; (void)ws_size;

    const v4f* z0l = (const v4f*)d_in[0];   // (N, 256)
    const v4f* z1l = (const v4f*)d_in[1];   // (N, 3, 256)
    const v4f* z0r = (const v4f*)d_in[2];   // (N, 256)
    const v4f* z1r = (const v4f*)d_in[3];   // (N, 3, 256)
    v4f* out       = (v4f*)d_out;           // (N, 3, 768)

    const int N     = in_sizes[0] / RANK;   // 50000
    const int total = N * R4;               // one thread per 16B rank chunk

    const int block = 256;                  // 8 wave32s
    const int grid  = (total + block - 1) / block;

    etp_o1_stream_kernel<<<grid, block, 0, stream>>>(z0l, z1l, z0r, z1r, out, total);
}